// HistoryFilterLinear_30631706755830
// MI455X (gfx1250) — compile-verified
//
#include <hip/hip_runtime.h>
#include <hip/hip_bf16.h>

#define N_S   50000
#define N_A   50000
#define E_A   800000
#define E_S   800000
#define HID   128

// Aggregation workspace strides (K-padded to multiples of 4 for WMMA)
#define STR_U 24    // [u(16), posA(2), deg*posDst(2)@node, dis(1)@20, deg@21, pad]
#define STR_X 168   // [x(32), h(128), posSrc(2), deg*posDst(2)@node, dis@164, deg@165, pad]
#define STR_3 420   // physical: [h(0:128), sum_u(128:256), mean_x(256:384), x(384:416), pos(416:418), pad]
#define K_U   21
#define K_X   165

typedef float v2f __attribute__((ext_vector_type(2)));
typedef float v8f __attribute__((ext_vector_type(8)));
typedef int   v4i __attribute__((ext_vector_type(4)));

// ---- CDNA5 async global->LDS path (builtin confirmed present; v4i-typed) ----
#if __has_builtin(__builtin_amdgcn_global_load_async_to_lds_b128)
#define HAVE_ASYNC 1
#define ASYNC_CP16(lp, gp)                                        \
    __builtin_amdgcn_global_load_async_to_lds_b128(               \
        (__attribute__((address_space(1))) v4i*)(gp),             \
        (__attribute__((address_space(3))) v4i*)(lp), 0, 0)
#else
#define HAVE_ASYNC 0
#endif

#if __has_builtin(__builtin_amdgcn_s_wait_asynccnt)
#define ASYNC_WAIT() __builtin_amdgcn_s_wait_asynccnt(0)
#else
#define ASYNC_WAIT() asm volatile("s_wait_asynccnt 0" ::: "memory")
#endif

// ---------------------------------------------------------------- zero ws
__global__ __launch_bounds__(256) void zero_ws_kernel(float* __restrict__ p, int n) {
    int i = blockIdx.x * 256 + threadIdx.x;
    if (i < n) p[i] = 0.0f;
}

// ------------------------------------------------------- a2s edge scatter
__global__ __launch_bounds__(256) void agg_a2s_kernel(
    const float* __restrict__ u, const float* __restrict__ pos_action,
    const float* __restrict__ dis, const int* __restrict__ src,
    const int* __restrict__ dst, float* __restrict__ aggU) {
    int wave = (blockIdx.x * 256 + threadIdx.x) >> 5;
    int lane = threadIdx.x & 31;
    if (wave >= E_A) return;
    int s = src[wave], d = dst[wave];
    float val = 0.0f; int slot = -1;
    if (lane < 16)       { val = u[s * 16 + lane];              slot = lane; }
    else if (lane < 18)  { val = pos_action[s * 2 + lane - 16]; slot = lane; }
    else if (lane == 18) { val = dis[wave];                     slot = 20;   }
    else if (lane == 19) { val = 1.0f;                          slot = 21;   }
    if (slot >= 0) atomicAdd(&aggU[(size_t)d * STR_U + slot], val);
}

// ------------------------------------------------------- s2s edge scatter
__global__ __launch_bounds__(256) void agg_s2s_kernel(
    const float* __restrict__ x, const float* __restrict__ h,
    const float* __restrict__ pos_state, const float* __restrict__ dis,
    const int* __restrict__ src, const int* __restrict__ dst,
    float* __restrict__ aggX) {
    int wave = (blockIdx.x * 256 + threadIdx.x) >> 5;
    int lane = threadIdx.x & 31;
    if (wave >= E_S) return;
    int s = src[wave], d = dst[wave];
#pragma unroll
    for (int j = 0; j < 6; ++j) {
        int f = lane + 32 * j;
        float val; bool wr = true;
        if (f < 32)        val = x[(size_t)s * 32 + f];
        else if (f < 160)  val = h[(size_t)s * HID + (f - 32)];
        else if (f < 162)  val = pos_state[(size_t)s * 2 + (f - 160)];
        else if (f == 164) val = dis[wave];
        else if (f == 165) val = 1.0f;          // degree
        else wr = false;                        // 162,163 (dst terms), 166+
        if (wr) atomicAdd(&aggX[(size_t)d * STR_X + f], val);
    }
}

// --------------------------------------------------------- WMMA GEMM tiles
template <int KT, int STRIDE, int UNROLL>
__device__ inline v8f gemm_tile(const float* __restrict__ As,
                                const float* __restrict__ W, int Kin,
                                int n0, int lane) {
    v8f acc = {};
    int m  = lane & 15;
    int kh = (lane >> 4) << 1;        // 0 or 2
#pragma unroll UNROLL
    for (int kt = 0; kt < KT; ++kt) {
        int kb = 4 * kt + kh;
        v2f a, b;
        a.x = As[m * STRIDE + kb];
        a.y = As[m * STRIDE + kb + 1];
        b.x = (kb     < Kin) ? W[(size_t)kb       * HID + n0 + m] : 0.0f;
        b.y = (kb + 1 < Kin) ? W[(size_t)(kb + 1) * HID + n0 + m] : 0.0f;
        acc = __builtin_amdgcn_wmma_f32_16x16x4_f32(
            false, a, false, b, (short)0, acc, false, false);
    }
    return acc;
}

// GEMM3 uses the rotated A3 layout: physical col k -> W_upd row (k+2), pos tail -> rows 0,1
__device__ inline v8f gemm_tile_rot(const float* __restrict__ As,
                                    const float* __restrict__ W,
                                    int n0, int lane) {
    v8f acc = {};
    int m  = lane & 15;
    int kh = (lane >> 4) << 1;
#pragma unroll 5
    for (int kt = 0; kt < STR_3 / 4; ++kt) {
        int kb = 4 * kt + kh;
        int r0 = (kb     < 416) ? kb + 2 : ((kb     < 418) ? kb - 416 : -1);
        int r1 = (kb + 1 < 416) ? kb + 3 : ((kb + 1 < 418) ? kb - 415 : -1);
        v2f a, b;
        a.x = As[m * STR_3 + kb];
        a.y = As[m * STR_3 + kb + 1];
        b.x = (r0 >= 0) ? W[(size_t)r0 * HID + n0 + m] : 0.0f;
        b.y = (r1 >= 0) ? W[(size_t)r1 * HID + n0 + m] : 0.0f;
        acc = __builtin_amdgcn_wmma_f32_16x16x4_f32(
            false, a, false, b, (short)0, acc, false, false);
    }
    return acc;
}

// --------------------------------------------------------- node update
__global__ __launch_bounds__(256) void node_update_kernel(
    const float* __restrict__ h, const float* __restrict__ x,
    const float* __restrict__ pos_state,
    const float* __restrict__ aggU, const float* __restrict__ aggX,
    const float* __restrict__ W_u2h, const float* __restrict__ b_u2h,
    const float* __restrict__ W_x2h, const float* __restrict__ b_x2h,
    const float* __restrict__ W_upd, const float* __restrict__ b_upd,
    float* __restrict__ out) {
    __shared__ float A1[16 * STR_U];       // raw aggU rows (+fixups)
    __shared__ float A2[16 * STR_X];       // raw aggX rows (+fixups)
    __shared__ float A3[16 * STR_3];       // [h | sum_u | mean_x | x | pos | pad]
    __shared__ float sdegU[16], sdegS[16];

    const int tid   = threadIdx.x;
    const int node0 = blockIdx.x * 16;
    const int lane  = tid & 31;
    const int wv    = tid >> 5;            // 0..7

    // L2 prefetch of the big weight matrix (L2-resident across all blocks)
    for (int ofs = tid * 32; ofs < 418 * HID; ofs += 256 * 32)
        __builtin_prefetch(W_upd + ofs, 0, 1);

#if HAVE_ASYNC
    // aggU block: 16*24 floats contiguous = 3 x 512B chunks (wave 0)
    if (wv == 0) {
        const float* g = aggU + (size_t)node0 * STR_U;
#pragma unroll
        for (int c = 0; c < 3; ++c)
            ASYNC_CP16(&A1[c * 128 + lane * 4], g + c * 128 + lane * 4);
    } else {
        // aggX block: 16*168 floats contiguous = 21 x 512B chunks (waves 1..7)
        const float* g = aggX + (size_t)node0 * STR_X;
        for (int c = wv - 1; c < 21; c += 7)
            ASYNC_CP16(&A2[c * 128 + lane * 4], g + c * 128 + lane * 4);
    }
    // h rows -> A3 cols 0..127 (512B per row; 2 rows per wave)
#pragma unroll
    for (int r = wv; r < 16; r += 8)
        ASYNC_CP16(&A3[r * STR_3 + lane * 4],
                   h + (size_t)(node0 + r) * HID + lane * 4);
    // x rows -> A3 cols 384..415 (128B per row; lanes 0..15 cover 2 rows)
    if (lane < 16) {
        int r  = wv * 2 + (lane >> 3);
        int l8 = lane & 7;
        ASYNC_CP16(&A3[r * STR_3 + 384 + l8 * 4],
                   x + (size_t)(node0 + r) * 32 + l8 * 4);
    }
    // pos -> A3 cols 416..417, zero pad cols 418..419 (plain DS stores)
    if (tid < 32) {
        int m = tid >> 1, c = tid & 1;
        A3[m * STR_3 + 416 + c] = pos_state[(size_t)(node0 + m) * 2 + c];
        A3[m * STR_3 + 418 + c] = 0.0f;
    }
    ASYNC_WAIT();
    __syncthreads();
#else
    // synchronous fallback: raw copies into LDS
    for (int i = tid; i < 16 * STR_U; i += 256)
        A1[i] = aggU[(size_t)node0 * STR_U + i];
    for (int i = tid; i < 16 * STR_X; i += 256)
        A2[i] = aggX[(size_t)node0 * STR_X + i];
    for (int i = tid; i < 16 * STR_3; i += 256) {
        int m = i / STR_3, c = i % STR_3;
        size_t nd = node0 + m;
        if (c < 128)        A3[i] = h[nd * HID + c];
        else if (c < 384)   continue;                     // filled by GEMM1/2
        else if (c < 416)   A3[i] = x[nd * 32 + (c - 384)];
        else if (c < 418)   A3[i] = pos_state[nd * 2 + (c - 416)];
        else                A3[i] = 0.0f;
    }
    __syncthreads();
#endif

    // LDS-side fixups: deg*pos_dst terms + degree extraction
    if (tid < 32) {
        int m = tid >> 1, c = tid & 1;
        float p = A3[m * STR_3 + 416 + c];                 // pos_state[node][c]
        A1[m * STR_U + 18 + c]  = A1[m * STR_U + 21] * p;  // deg_u * pos_dst
        A2[m * STR_X + 162 + c] = A2[m * STR_X + 165] * p; // deg_s * pos_dst
    }
    if (tid < 16) {
        sdegU[tid] = A1[tid * STR_U + 21];
        sdegS[tid] = A2[tid * STR_X + 165];
    }
    __syncthreads();

    const int n0 = wv * 16;
    const int cn = lane & 15;
    const int mb = (lane >> 4) * 8;

    // GEMM1: sum_u = A1(16x21) @ W_u2h + degU*b  -> A3 cols 128..255
    {
        v8f acc = gemm_tile<STR_U / 4, STR_U, 6>(A1, W_u2h, K_U, n0, lane);
#pragma unroll
        for (int v = 0; v < 8; ++v) {
            int m = mb + v;
            A3[m * STR_3 + 128 + n0 + cn] = acc[v] + sdegU[m] * b_u2h[n0 + cn];
        }
    }
    // GEMM2: mean_x = (A2(16x165) @ W_x2h + degS*b)/degS -> A3 cols 256..383
    {
        v8f acc = gemm_tile<STR_X / 4, STR_X, 6>(A2, W_x2h, K_X, n0, lane);
#pragma unroll
        for (int v = 0; v < 8; ++v) {
            int m  = mb + v;
            float dg = sdegS[m];
            float r  = acc[v] + dg * b_x2h[n0 + cn];
            A3[m * STR_3 + 256 + n0 + cn] = (dg > 0.0f) ? (r / dg) : 0.0f;
        }
    }
    __syncthreads();

    // GEMM3: out = A3(16x418, rotated) @ W_upd + b_upd
    {
        v8f acc = gemm_tile_rot(A3, W_upd, n0, lane);
#pragma unroll
        for (int v = 0; v < 8; ++v) {
            int m = mb + v;
            out[(size_t)(node0 + m) * HID + n0 + cn] = acc[v] + b_upd[n0 + cn];
        }
    }
}

// ---------------------------------------------------------------- launch
extern "C" void kernel_launch(void* const* d_in, const int* in_sizes, int n_in,
                              void* d_out, int out_size, void* d_ws, size_t ws_size,
                              hipStream_t stream) {
    const float* h          = (const float*)d_in[0];
    const float* x          = (const float*)d_in[1];
    const float* u          = (const float*)d_in[2];
    const float* pos_state  = (const float*)d_in[3];
    const float* pos_action = (const float*)d_in[4];
    const float* dis_a2s    = (const float*)d_in[5];
    const float* dis_s2s    = (const float*)d_in[6];
    const int*   a2s_src    = (const int*)d_in[7];
    const int*   a2s_dst    = (const int*)d_in[8];
    const int*   s2s_src    = (const int*)d_in[9];
    const int*   s2s_dst    = (const int*)d_in[10];
    const float* W_u2h      = (const float*)d_in[11];
    const float* b_u2h      = (const float*)d_in[12];
    const float* W_x2h      = (const float*)d_in[13];
    const float* b_x2h      = (const float*)d_in[14];
    const float* W_upd      = (const float*)d_in[15];
    const float* b_upd      = (const float*)d_in[16];
    float* out = (float*)d_out;

    float* aggU = (float*)d_ws;                       // N_S * 24
    float* aggX = aggU + (size_t)N_S * STR_U;         // N_S * 168
    const int totz = N_S * (STR_U + STR_X);

    zero_ws_kernel<<<(totz + 255) / 256, 256, 0, stream>>>(aggU, totz);
    agg_a2s_kernel<<<(E_A + 7) / 8, 256, 0, stream>>>(u, pos_action, dis_a2s,
                                                      a2s_src, a2s_dst, aggU);
    agg_s2s_kernel<<<(E_S + 7) / 8, 256, 0, stream>>>(x, h, pos_state, dis_s2s,
                                                      s2s_src, s2s_dst, aggX);
    node_update_kernel<<<N_S / 16, 256, 0, stream>>>(h, x, pos_state, aggU, aggX,
                                                     W_u2h, b_u2h, W_x2h, b_x2h,
                                                     W_upd, b_upd, out);
}